// YOLOv2Loss_69483980915392
// MI455X (gfx1250) — compile-verified
//
#include <hip/hip_runtime.h>
#include <math.h>

// ---------------------------------------------------------------------------
// YOLOv2 loss for MI455X (gfx1250, wave32).
// Memory-bound: ~300-400MB traffic -> bandwidth-limited; all hot loops are
// coalesced along hw (lane stride = 4B). Reductions are deterministic
// (block partials -> single final block) and use V_WMMA_F32_16X16X4_F32
// with an all-ones A matrix as the intra-wave sum primitive (exact f32).
// ---------------------------------------------------------------------------

typedef float v2f __attribute__((ext_vector_type(2)));
typedef float v8f __attribute__((ext_vector_type(8)));

#define NB   64
#define NA   5
#define WD   52
#define HD   52
#define HWD  (WD * HD)          // 2704
#define NM   50
#define NCLS 80
#define NCH  (5 + NCLS)         // 85
#define NROW (NB * NA * HWD)    // 865280
#define TPB  256
#define NBLK (NROW / TPB)       // 3380 (exact)

__device__ __forceinline__ float sigmoidf_(float x) { return 1.0f / (1.0f + expf(-x)); }

__device__ __forceinline__ float iou_xywh_dev(float ax, float ay, float aw, float ah,
                                              float bx, float by, float bw, float bh,
                                              float area_b) {
  float tlx = fmaxf(ax - aw * 0.5f, bx - bw * 0.5f);
  float tly = fmaxf(ay - ah * 0.5f, by - bh * 0.5f);
  float brx = fminf(ax + aw * 0.5f, bx + bw * 0.5f);
  float bry = fminf(ay + ah * 0.5f, by + bh * 0.5f);
  float en  = ((tlx < brx) && (tly < bry)) ? 1.0f : 0.0f;
  float ai  = (brx - tlx) * (bry - tly) * en;
  return ai / (aw * ah + area_b - ai + 1e-12f);
}

// Block reduction. 256 threads -> 8-way LDS fold into wave 0 (32 partials),
// then one V_WMMA_F32_16X16X4_F32 with A = ones(16x4): D[0][n] = column sums
// of B, so sum over lanes 0..15 of d[0] == sum of all 64 B elements (the 32
// partials + 32 zeros), independent of B's lane striping. Exact f32 math.
// EXEC is all-ones for wave 0 inside the tid<32 region (WMMA requirement).
__device__ __forceinline__ void block_reduce_wmma(float local, float* dst, float scale) {
  __shared__ float red[TPB];
  int tid = threadIdx.x;
  red[tid] = local;
  __syncthreads();
  if (tid < 32) {
    float p = 0.0f;
#pragma unroll
    for (int i = 0; i < 8; ++i) p += red[tid + 32 * i];
    v2f a;  a[0] = 1.0f; a[1] = 1.0f;   // A-matrix: all ones (16x4 f32)
    v2f bb; bb[0] = p;   bb[1] = 0.0f;  // B-matrix: partials + zeros (4x16 f32)
    v8f c = {};
    v8f d = __builtin_amdgcn_wmma_f32_16x16x4_f32(false, a, false, bb,
                                                  (short)0, c, false, false);
    red[tid] = d[0];                    // row M=0 of D on lanes 0..15
  }
  __syncthreads();
  if (tid == 0) {
    float s = 0.0f;
#pragma unroll
    for (int i = 0; i < 16; ++i) s += red[i];
    dst[0] = s * scale;
  }
}

// --- Kernel 0: per-batch num_obj count; zero ignore flags --------------------
__global__ void k_init(const float* __restrict__ tgt, int* __restrict__ num_obj,
                       int* __restrict__ flags) {
  int b = threadIdx.x;
  if (b < NB) {
    int cnt = 0;
    for (int m = 0; m < NM; ++m) {
      const float* p = tgt + (b * NM + m) * 5;
      float s = p[0] + p[1] + p[2] + p[3] + p[4];
      if (s > 0.0f) ++cnt;
    }
    num_obj[b] = cnt;
    flags[b]   = 0;
  }
}

// --- Kernel 1: pred boxes (b,a,hw layout; coalesced along hw) ----------------
__global__ void __launch_bounds__(TPB) k_pred(const float* __restrict__ outp,
                                              const float* __restrict__ anch,
                                              float* __restrict__ pred) {
  int t  = blockIdx.x * TPB + threadIdx.x;
  int b  = t / (NA * HWD);
  int r  = t - b * (NA * HWD);
  int a  = r / HWD;
  int hw = r - a * HWD;
  int x  = hw % WD, y = hw / WD;
  const float* base = outp + ((b * NA + a) * NCH) * HWD + hw;
  float tx = base[0];
  float ty = base[HWD];
  float tw = base[2 * HWD];
  float th = base[3 * HWD];
  float aw = anch[a * 2 + 0], ah = anch[a * 2 + 1];
  float bx = fminf(fmaxf(sigmoidf_(tx) + (float)x, 0.0f), (float)WD);
  float by = fminf(fmaxf(sigmoidf_(ty) + (float)y, 0.0f), (float)HD);
  float bw = fminf(fmaxf(expf(tw) * aw, 0.0f), (float)WD);
  float bh = fminf(fmaxf(expf(th) * ah, 0.0f), (float)HD);
  pred[t * 4 + 0] = bx;
  pred[t * 4 + 1] = by;
  pred[t * 4 + 2] = bw;
  pred[t * 4 + 3] = bh;
}

// --- Kernel 2: max IoU of each pred box vs valid GT; batch ignore flag -------
__global__ void __launch_bounds__(TPB) k_maxiou(const float* __restrict__ tgt,
                                                const float* __restrict__ pred,
                                                const int* __restrict__ num_obj,
                                                float* __restrict__ max_iou,
                                                int* __restrict__ flags) {
  __shared__ float g[NM][5];  // gx, gy, gw, gh, area
  __shared__ int   bf;
  int b   = blockIdx.y;
  int tid = threadIdx.x;
  if (tid == 0) bf = 0;
  if (tid < NM) {
    const float* p = tgt + (b * NM + tid) * 5;
    float gx = p[1] * (float)WD, gy = p[2] * (float)HD;
    float gw = p[3] * (float)WD, gh = p[4] * (float)HD;
    g[tid][0] = gx; g[tid][1] = gy; g[tid][2] = gw; g[tid][3] = gh;
    g[tid][4] = gw * gh;
  }
  __syncthreads();
  int idx = blockIdx.x * TPB + tid;
  int no  = num_obj[b];
  if (idx < NA * HWD) {
    int   t  = b * (NA * HWD) + idx;
    float px = pred[t * 4 + 0], py = pred[t * 4 + 1];
    float pw = pred[t * 4 + 2], ph = pred[t * 4 + 3];
    float miou = -1.0f;  // invalid GT rows masked to -1 in reference
    for (int m = 0; m < no; ++m) {
      float v = iou_xywh_dev(px, py, pw, ph, g[m][0], g[m][1], g[m][2], g[m][3], g[m][4]);
      miou = fmaxf(miou, v);
    }
    max_iou[t] = miou;
    if (miou > 0.5f) bf = 1;  // apply_ign uses strict >
  }
  __syncthreads();
  if (tid == 0 && bf) atomicOr(&flags[b], 1);
}

// --- Kernel 3: target-array init (noobj mask from ignore flag) ---------------
__global__ void __launch_bounds__(TPB) k_tgtinit(const float* __restrict__ max_iou,
                                                 const int* __restrict__ num_obj,
                                                 const int* __restrict__ flags,
                                                 float* __restrict__ iou_t,
                                                 float* __restrict__ iou_m,
                                                 float* __restrict__ box_t,
                                                 float* __restrict__ bmbs,
                                                 float* __restrict__ cls_t,
                                                 float* __restrict__ cls_m) {
  int t = blockIdx.x * TPB + threadIdx.x;
  int b = t / (NA * HWD);
  float im = 0.0f;
  if (num_obj[b] > 0)
    im = (flags[b] != 0 && max_iou[t] >= 0.5f) ? 0.0f : 1.0f;  // mask uses >=
  iou_t[t] = 0.0f;
  iou_m[t] = im;
  box_t[t * 4 + 0] = 0.0f; box_t[t * 4 + 1] = 0.0f;
  box_t[t * 4 + 2] = 0.0f; box_t[t * 4 + 3] = 0.0f;
  bmbs[t] = 0.0f;
  cls_t[t] = 0.0f;
  cls_m[t] = 0.0f;
}

// --- Kernel 4: sequential per-batch assignment scan (last writer wins) -------
__global__ void k_assign(const float* __restrict__ tgt, const float* __restrict__ anch,
                         const float* __restrict__ pred, const float* __restrict__ max_iou,
                         const int* __restrict__ num_obj,
                         float* __restrict__ iou_t, float* __restrict__ iou_m,
                         float* __restrict__ box_t, float* __restrict__ bmbs,
                         float* __restrict__ cls_t, float* __restrict__ cls_m) {
  int b = threadIdx.x;
  if (b >= NB) return;
  float aw[NA], ah[NA];
#pragma unroll
  for (int a = 0; a < NA; ++a) { aw[a] = anch[a * 2]; ah[a] = anch[a * 2 + 1]; }
  int no = num_obj[b];
  for (int m = 0; m < no; ++m) {  // sequential, matches lax.scan ordering
    const float* p   = tgt + (b * NM + m) * 5;
    float gcls = p[0];
    float gx = p[1] * (float)WD, gy = p[2] * (float)HD;
    float gw = p[3] * (float)WD, gh = p[4] * (float)HD;
    int cx = (int)floorf(gx), cy = (int)floorf(gy);
    cx = cx < 0 ? 0 : (cx > WD - 1 ? WD - 1 : cx);
    cy = cy < 0 ? 0 : (cy > HD - 1 ? HD - 1 : cy);
    int   cell   = cy * WD + cx;
    float area_g = gw * gh;
    float best = -2.0f;
    int   bi   = 0;
#pragma unroll
    for (int a = 0; a < NA; ++a) {  // argmax, first occurrence on ties
      float ov = iou_xywh_dev((float)cx + 0.5f, (float)cy + 0.5f, aw[a], ah[a],
                              gx, gy, gw, gh, area_g);
      if (ov > best) { best = ov; bi = a; }
    }
    int   row = (b * NA + bi) * HWD + cell;
    float pw = pred[row * 4 + 2], ph = pred[row * 4 + 3];
    float s  = 2.0f - (pw / (float)WD) * (ph / (float)HD);
    iou_t[row] = max_iou[row];
    iou_m[row] = 2.0f;
    box_t[row * 4 + 0] = gx - (float)cx;
    box_t[row * 4 + 1] = gy - (float)cy;
    box_t[row * 4 + 2] = gw / aw[bi];
    box_t[row * 4 + 3] = gh / ah[bi];
    bmbs[row]  = s;        // box_m(=1) * box_s
    cls_t[row] = gcls;
    cls_m[row] = 1.0f;
  }
}

// --- Kernel 5: main loss pass (one coalesced sweep of outputs) ---------------
__global__ void __launch_bounds__(TPB) k_loss(const float* __restrict__ outp,
                                              const float* __restrict__ iou_t,
                                              const float* __restrict__ iou_m,
                                              const float* __restrict__ box_t,
                                              const float* __restrict__ bmbs,
                                              const float* __restrict__ cls_t,
                                              const float* __restrict__ cls_m,
                                              float* __restrict__ blockSums) {
  int t  = blockIdx.x * TPB + threadIdx.x;
  int b  = t / (NA * HWD);
  int r  = t - b * (NA * HWD);
  int a  = r / HWD;
  int hw = r - a * HWD;
  const float* base = outp + ((b * NA + a) * NCH) * HWD + hw;

  float x0 = base[0], x1 = base[HWD], x2 = base[2 * HWD], x3 = base[3 * HWD];
  float x4 = base[4 * HWD];
  float d0 = sigmoidf_(x0), d1 = sigmoidf_(x1);
  float d2 = expf(x2),      d3 = expf(x3);
  float conf = sigmoidf_(x4);

  float w   = bmbs[t];
  float wd0 = (d0 - box_t[t * 4 + 0]) * w;
  float wd1 = (d1 - box_t[t * 4 + 1]) * w;
  float wd2 = (d2 - box_t[t * 4 + 2]) * w;
  float wd3 = (d3 - box_t[t * 4 + 3]) * w;
  float sumw = wd0 * wd0 + wd1 * wd1 + wd2 * wd2 + wd3 * wd3;

  float imv   = iou_m[t];
  float cdiff = conf - iou_t[t];
  float wiou  = (imv == 1.0f) ? 1.0f : ((imv == 2.0f) ? 5.0f : 0.0f);
  float csq   = wiou * cdiff * cdiff;

  int ci = (int)cls_t[t];
  ci = ci < 0 ? 0 : (ci > NCLS - 1 ? NCLS - 1 : ci);
  float mx = -__builtin_inff(), se = 0.0f, xc = 0.0f;
  for (int ch = 0; ch < NCLS; ++ch) {  // online log-softmax: single pass
    float x = base[(5 + ch) * HWD];
    xc = (ch == ci) ? x : xc;
    float nm = fmaxf(mx, x);
    se = se * expf(mx - nm) + expf(x - nm);
    mx = nm;
  }
  float ce = -(xc - mx - logf(se));

  float local = 0.5f * sumw + 0.5f * csq + cls_m[t] * ce;
  block_reduce_wmma(local, &blockSums[blockIdx.x], 1.0f);
}

// --- Kernel 6: deterministic final reduction, /B scale -----------------------
__global__ void __launch_bounds__(TPB) k_final(const float* __restrict__ blockSums,
                                               float* __restrict__ out) {
  float local = 0.0f;
  for (int i = threadIdx.x; i < NBLK; i += TPB) local += blockSums[i];
  block_reduce_wmma(local, out, 1.0f / (float)NB);
}

// ---------------------------------------------------------------------------
extern "C" void kernel_launch(void* const* d_in, const int* in_sizes, int n_in,
                              void* d_out, int out_size, void* d_ws, size_t ws_size,
                              hipStream_t stream) {
  (void)in_sizes; (void)n_in; (void)out_size; (void)ws_size;
  const float* outputs = (const float*)d_in[0];  // (64,425,52,52) f32
  const float* targets = (const float*)d_in[1];  // (64,50,5) f32
  const float* anchors = (const float*)d_in[2];  // (5,2) f32

  // Workspace layout (floats), total ~48.5 MB.
  float* ws      = (float*)d_ws;
  float* pred    = ws;                                  // NROW*4
  float* max_iou = pred + (size_t)NROW * 4;             // NROW
  float* iou_t   = max_iou + NROW;                      // NROW
  float* iou_m   = iou_t + NROW;                        // NROW
  float* box_t   = iou_m + NROW;                        // NROW*4
  float* bmbs    = box_t + (size_t)NROW * 4;            // NROW
  float* cls_t   = bmbs + NROW;                         // NROW
  float* cls_m   = cls_t + NROW;                        // NROW
  float* bsums   = cls_m + NROW;                        // NBLK
  int*   num_obj = (int*)(bsums + NBLK);                // NB
  int*   flags   = num_obj + NB;                        // NB

  k_init<<<1, 64, 0, stream>>>(targets, num_obj, flags);
  k_pred<<<NROW / TPB, TPB, 0, stream>>>(outputs, anchors, pred);
  dim3 gB((NA * HWD + TPB - 1) / TPB, NB);
  k_maxiou<<<gB, TPB, 0, stream>>>(targets, pred, num_obj, max_iou, flags);
  k_tgtinit<<<NROW / TPB, TPB, 0, stream>>>(max_iou, num_obj, flags,
                                            iou_t, iou_m, box_t, bmbs, cls_t, cls_m);
  k_assign<<<1, 64, 0, stream>>>(targets, anchors, pred, max_iou, num_obj,
                                 iou_t, iou_m, box_t, bmbs, cls_t, cls_m);
  k_loss<<<NROW / TPB, TPB, 0, stream>>>(outputs, iou_t, iou_m, box_t, bmbs,
                                         cls_t, cls_m, bsums);
  k_final<<<1, TPB, 0, stream>>>(bsums, (float*)d_out);
}